// ConditionalBatchNorm1D_40785009443474
// MI455X (gfx1250) — compile-verified
//
#include <hip/hip_runtime.h>

#define NUM_COND 16
#define FEAT 128
#define CBN_EPS 1e-5f

typedef __attribute__((ext_vector_type(2))) float v2f;
typedef __attribute__((ext_vector_type(4))) float v4f;
typedef __attribute__((ext_vector_type(8))) float v8f;

// ---------------------------------------------------------------------------
// Kernel 0: zero the stats region of the workspace (cnt[16], s1[16*128],
// s2[16*128]) every call — harness does not re-poison between replays.
// ---------------------------------------------------------------------------
__global__ void cbn_zero(float* __restrict__ ws, int n) {
    int i = blockIdx.x * blockDim.x + threadIdx.x;
    if (i < n) ws[i] = 0.0f;
}

// ---------------------------------------------------------------------------
// Kernel 1: segmented sums via one-hot fp32 WMMA.
//   s1[c][f] = sum_{n: labels[n]==c} x[n][f]
//   s2[c][f] = sum_{n: labels[n]==c} x[n][f]^2
//   cnt[c]   = #{n: labels[n]==c}
// One wave owns one 16-feature tile (8 waves x 16 = 128 features).
// Fast path folds 8 rows/iteration with four V_WMMA_F32_16X16X4_F32
// (no predication); a wave-uniform tail path handles N%8 rows.
//   A (16x4)  = one-hot of 4 labels (M = condition = lane%16)
//   B (4x16)  = the 4 x-rows (N = feature = lane%16)  /  B^2 for s2
// ISA layouts (7.12.2): A lanes0-15 hold K=0,1; lanes16-31 hold K=2,3.
// D VGPR j: M = j + 8*(lane>=16), N = lane%16.
// ---------------------------------------------------------------------------
__global__ __launch_bounds__(256) void cbn_stats_wmma(
    const float* __restrict__ x, const int* __restrict__ labels,
    float* __restrict__ cnt, float* __restrict__ s1, float* __restrict__ s2,
    int N) {
    const int lane = threadIdx.x & 31;
    const int wave = threadIdx.x >> 5;   // feature tile 0..7
    const int lm   = lane & 15;          // M index (cond) / N index (feature)
    const int lg   = lane >> 4;          // 0 -> rows k=0,1 ; 1 -> rows k=2,3
    const int col  = wave * 16 + lm;     // this lane's feature column

    v8f acc1 = {};   // s1 accumulator tile
    v8f acc2 = {};   // s2 accumulator tile
    float cloc = 0.0f;

    const int stride = (int)gridDim.x * 8;
    for (int base = (int)blockIdx.x * 8; base < N; base += stride) {
        if (base + 8 <= N) {
            // -------- fast path: full 8-row chunk, no masks --------
            const int r0 = base + lg * 2;            // rows for this lane half
            // paired label loads (8-byte aligned: base%8==0, lg*2 even)
            const int2 lp0 = *(const int2*)(labels + r0);
            const int2 lp1 = *(const int2*)(labels + r0 + 4);
            const size_t xb = (size_t)r0 * FEAT + col;
            const float bx0 = x[xb];
            const float by0 = x[xb + FEAT];
            const float bx1 = x[xb + 4 * FEAT];
            const float by1 = x[xb + 5 * FEAT];

            v2f a0, a1, b0, b1, q0, q1;
            a0.x = (lp0.x == lm) ? 1.0f : 0.0f;
            a0.y = (lp0.y == lm) ? 1.0f : 0.0f;
            a1.x = (lp1.x == lm) ? 1.0f : 0.0f;
            a1.y = (lp1.y == lm) ? 1.0f : 0.0f;
            b0.x = bx0; b0.y = by0;
            b1.x = bx1; b1.y = by1;
            q0.x = bx0 * bx0; q0.y = by0 * by0;
            q1.x = bx1 * bx1; q1.y = by1 * by1;

            acc1 = __builtin_amdgcn_wmma_f32_16x16x4_f32(
                false, a0, false, b0, (short)0, acc1, false, false);
            acc2 = __builtin_amdgcn_wmma_f32_16x16x4_f32(
                false, a0, false, q0, (short)0, acc2, false, false);
            acc1 = __builtin_amdgcn_wmma_f32_16x16x4_f32(
                false, a1, false, b1, (short)0, acc1, false, false);
            acc2 = __builtin_amdgcn_wmma_f32_16x16x4_f32(
                false, a1, false, q1, (short)0, acc2, false, false);

            if (wave == 0) cloc += (a0.x + a0.y) + (a1.x + a1.y);

            // prefetch next grid-stride chunk (wave-uniform guard)
            if (base + stride + 8 <= N) {
                const size_t xn = (size_t)(base + stride + lg * 2) * FEAT + col;
                __builtin_prefetch(x + xn, 0, 1);
                __builtin_prefetch(x + xn + 4 * FEAT, 0, 1);
            }
        } else {
            // -------- tail path: predicated 4-row sub-chunks --------
            for (int s = 0; s < 8; s += 4) {
                const int r0 = base + s + lg * 2;
                const int r1 = r0 + 1;
                const bool v0 = r0 < N;
                const bool v1 = r1 < N;
                const int rc0 = v0 ? r0 : (N - 1);
                const int rc1 = v1 ? r1 : (N - 1);
                const int l0 = labels[rc0];
                const int l1 = labels[rc1];
                const float bx = x[(size_t)rc0 * FEAT + col];
                const float by = x[(size_t)rc1 * FEAT + col];

                v2f a, b, q;
                a.x = (v0 && l0 == lm) ? 1.0f : 0.0f;
                a.y = (v1 && l1 == lm) ? 1.0f : 0.0f;
                b.x = v0 ? bx : 0.0f;
                b.y = v1 ? by : 0.0f;
                q.x = b.x * b.x;
                q.y = b.y * b.y;

                acc1 = __builtin_amdgcn_wmma_f32_16x16x4_f32(
                    false, a, false, b, (short)0, acc1, false, false);
                acc2 = __builtin_amdgcn_wmma_f32_16x16x4_f32(
                    false, a, false, q, (short)0, acc2, false, false);

                if (wave == 0) cloc += a.x + a.y;
            }
        }
    }

    // Merge this block's partial tiles into global stats.
#pragma unroll
    for (int j = 0; j < 8; ++j) {
        const int m   = j + lg * 8;
        const int idx = m * FEAT + col;
        atomicAdd(&s1[idx], acc1[j]);
        atomicAdd(&s2[idx], acc2[j]);
    }
    if (wave == 0) atomicAdd(&cnt[lm], cloc);
}

// ---------------------------------------------------------------------------
// Kernel 2: finalize 16x128 stats -> scale/shift tables.
// ---------------------------------------------------------------------------
__global__ void cbn_finalize(const float* __restrict__ cnt,
                             const float* __restrict__ s1,
                             const float* __restrict__ s2,
                             const float* __restrict__ gamma,
                             const float* __restrict__ beta,
                             float* __restrict__ scale,
                             float* __restrict__ shift) {
    int i = blockIdx.x * blockDim.x + threadIdx.x;
    if (i >= NUM_COND * FEAT) return;
    const int c = i / FEAT;
    const float n    = fmaxf(cnt[c], 1.0f);
    const float mean = s1[i] / n;
    const float var  = s2[i] / n - mean * mean;
    const float inv  = rsqrtf(var + CBN_EPS);
    const float sc   = gamma[i] * inv;
    scale[i] = sc;
    shift[i] = beta[i] - mean * sc;
}

// ---------------------------------------------------------------------------
// Kernel 3: bandwidth-bound apply pass (dominates: ~512 MB of the ~768 MB
// total traffic). Stage 16 KB scale/shift in LDS, stream x -> y as v4f.
// x is dead after this read and y is never re-read: use non-temporal
// load/store so the 192 MB L2 isn't churned; prefetch one grid-stride ahead.
// ---------------------------------------------------------------------------
__global__ __launch_bounds__(256) void cbn_apply(
    const float* __restrict__ x, const int* __restrict__ labels,
    const float* __restrict__ scale, const float* __restrict__ shift,
    float* __restrict__ y, int N) {
    __shared__ v4f s_scale[NUM_COND * FEAT / 4];
    __shared__ v4f s_shift[NUM_COND * FEAT / 4];
    const v4f* g_sc = (const v4f*)scale;
    const v4f* g_sh = (const v4f*)shift;
    for (int i = threadIdx.x; i < NUM_COND * FEAT / 4; i += blockDim.x) {
        s_scale[i] = g_sc[i];
        s_shift[i] = g_sh[i];
    }
    __syncthreads();

    const v4f* x4 = (const v4f*)x;
    v4f* y4 = (v4f*)y;
    const long total  = (long)N * (FEAT / 4);
    const long stride = (long)gridDim.x * blockDim.x;
    for (long i = (long)blockIdx.x * blockDim.x + threadIdx.x; i < total;
         i += stride) {
        if (i + stride < total) __builtin_prefetch(&x4[i + stride], 0, 0);
        const long row = i >> 5;        // FEAT/4 == 32
        const int  c4  = (int)(i & 31);
        const int  lab = labels[row];
        const v4f xv = __builtin_nontemporal_load(&x4[i]);
        const v4f sc = s_scale[lab * 32 + c4];
        const v4f sh = s_shift[lab * 32 + c4];
        v4f o;
        o.x = fmaf(xv.x, sc.x, sh.x);
        o.y = fmaf(xv.y, sc.y, sh.y);
        o.z = fmaf(xv.z, sc.z, sh.z);
        o.w = fmaf(xv.w, sc.w, sh.w);
        __builtin_nontemporal_store(o, &y4[i]);
    }
}

// ---------------------------------------------------------------------------
// Launch
// ---------------------------------------------------------------------------
extern "C" void kernel_launch(void* const* d_in, const int* in_sizes, int n_in,
                              void* d_out, int out_size, void* d_ws, size_t ws_size,
                              hipStream_t stream) {
    const float* x      = (const float*)d_in[0];
    const int*   labels = (const int*)d_in[1];
    const float* gamma  = (const float*)d_in[2];
    const float* beta   = (const float*)d_in[3];
    float*       out    = (float*)d_out;
    const int N = in_sizes[1];   // labels: one per row

    // Workspace layout (floats): cnt[16] | s1[2048] | s2[2048] | scale[2048] | shift[2048]
    float* ws    = (float*)d_ws;
    float* cnt   = ws;
    float* s1    = ws + NUM_COND;
    float* s2    = s1 + NUM_COND * FEAT;
    float* scale = s2 + NUM_COND * FEAT;
    float* shift = scale + NUM_COND * FEAT;

    const int zn = NUM_COND + 2 * NUM_COND * FEAT;   // stats region
    cbn_zero<<<(zn + 255) / 256, 256, 0, stream>>>(ws, zn);

    cbn_stats_wmma<<<512, 256, 0, stream>>>(x, labels, cnt, s1, s2, N);

    cbn_finalize<<<(NUM_COND * FEAT + 255) / 256, 256, 0, stream>>>(
        cnt, s1, s2, gamma, beta, scale, shift);

    long totalVec = (long)N * (FEAT / 4);
    long nb = (totalVec + 255) / 256;
    if (nb > 2048) nb = 2048;
    if (nb < 1) nb = 1;
    cbn_apply<<<(int)nb, 256, 0, stream>>>(x, labels, scale, shift, out, N);
}